// LowRankOrthogonalProjection_39874476376601
// MI455X (gfx1250) — compile-verified
//
#include <hip/hip_runtime.h>
#include <stdint.h>

typedef __attribute__((ext_vector_type(2))) float v2f;
typedef __attribute__((ext_vector_type(8))) float v8f;

#define D_DIM  4096
#define TILE_M 16
#define CHUNK  256
#define NWAVES 4
#define TS     4100   // target tile LDS stride: 16400B rows (16B aligned); 4100%64=4 -> banks 4*m+c, conflict-free
#define SSY    260    // source staging stride: 1040B rows (16B aligned); 260%64=4 -> conflict-free
#define YS     17     // padded Y stride

#define SRC_OFF_BYTES (TILE_M * TS * 4)          // source staging starts after target tile

// out[row,:] = target[row,:] + ((source-target)[row,:] @ W) @ W^T
// Phase 1: Y = diff @ W via v_wmma_f32_16x16x4_f32, inputs streamed into LDS with
//          double-buffered global_load_async_to_lds_b128 (ASYNCcnt) overlapping WMMA.
// Phase 2: out = target(LDS) + Y @ W^T, 4 chained WMMAs per 16x16 block.
__global__ __launch_bounds__(128)
void lrproj_fused(const float* __restrict__ src,
                  const float* __restrict__ tgt,
                  const float* __restrict__ wgt,
                  float* __restrict__ out)
{
    extern __shared__ float smem[];
    float* tgtS  = smem;                         // TILE_M x TS (full target tile, async-filled)
    float* srcS  = smem + TILE_M * TS;           // 2 x TILE_M x SSY (double-buffered source chunk)
    float* yS    = srcS + 2 * TILE_M * SSY;      // TILE_M x YS
    float* partS = yS + TILE_M * YS;             // NWAVES x 256

    const int tid  = threadIdx.x;
    const int wave = tid >> 5;
    const int lane = tid & 31;
    const int half = lane >> 4;     // 0: lanes 0-15, 1: lanes 16-31
    const int l16  = lane & 15;
    const int klo  = half << 1;     // K offset per ISA f32 A/B layout

    const long row0 = (long)blockIdx.x * TILE_M;
    const float* __restrict__ srcT = src + row0 * D_DIM;
    const float* __restrict__ tgtT = tgt + row0 * D_DIM;
    float*       __restrict__ outT = out + row0 * D_DIM;

    const int cg = tid & 63;   // float4 column group within 256-col chunk
    const int rr = tid >> 6;   // row offset (0..1) for cooperative async issue

    // Issue one chunk's DMA: 16 async B128 per thread (8 rows x {src,tgt}).
    auto issue_chunk = [&](int c0, int pb) {
        const int col = cg << 2;
        #pragma unroll
        for (int m = rr; m < TILE_M; m += 2) {
            const uint32_t soff =
                (uint32_t)(SRC_OFF_BYTES + (pb * TILE_M * SSY + m * SSY + col) * 4);
            const uint32_t toff = (uint32_t)((m * TS + c0 + col) * 4);
            const float* sg = srcT + (long)m * D_DIM + c0 + col;
            const float* tg = tgtT + (long)m * D_DIM + c0 + col;
            asm volatile("global_load_async_to_lds_b128 %0, %1, off"
                         :: "v"(soff), "v"(sg) : "memory");
            asm volatile("global_load_async_to_lds_b128 %0, %1, off"
                         :: "v"(toff), "v"(tg) : "memory");
        }
    };

    v8f acc = {};   // per-wave partial Y (16x16 C/D layout)
    int pb = 0;

    // Prologue: fill buffer 0 / first target slab.
    issue_chunk(0, 0);
    asm volatile("s_wait_asynccnt 0x0" ::: "memory");
    __syncthreads();

    // ---------------- Phase 1: Y = (source - target) @ W ----------------
    for (int c0 = 0; c0 < D_DIM; c0 += CHUNK) {
        if (c0 + CHUNK < D_DIM) issue_chunk(c0 + CHUNK, pb ^ 1);  // overlap DMA with WMMA

        const float* sbuf = srcS + pb * TILE_M * SSY + l16 * SSY;
        const float* trow = tgtS + l16 * TS + c0;
        const int cw = wave << 6;          // this wave's 64 columns of the chunk
        #pragma unroll
        for (int kk = 0; kk < 64; kk += 4) {
            const int cc = cw + kk;
            const int d  = c0 + cc;
            // A (16x4): diff computed at load time from the two LDS copies
            const v2f s2 = *(const v2f*)(sbuf + cc + klo);
            const v2f t2 = *(const v2f*)(trow + cc + klo);
            const v2f a  = s2 - t2;
            // B (4x16): K rows of W (row-major D x 16), N = l16; L2-resident
            v2f b;
            b.x = wgt[(d + klo)     * 16 + l16];
            b.y = wgt[(d + klo + 1) * 16 + l16];
            acc = __builtin_amdgcn_wmma_f32_16x16x4_f32(
                false, a, false, b, (short)0, acc, false, false);
        }

        asm volatile("s_wait_asynccnt 0x0" ::: "memory");  // next chunk landed
        __syncthreads();
        pb ^= 1;
    }

    // Cross-wave reduction of partial Y through LDS.
    #pragma unroll
    for (int i = 0; i < 8; ++i)
        partS[(wave << 8) + (lane << 3) + i] = acc[i];
    __syncthreads();

    for (int e = tid; e < 256; e += 128) {
        float s = 0.f;
        #pragma unroll
        for (int w = 0; w < NWAVES; ++w) s += partS[(w << 8) + e];
        const int n = (e >> 3) & 15;                 // N = lane&15
        const int m = (e & 7) + ((e >> 7) << 3);     // M = vgpr + 8*(lane>=16)
        yS[m * YS + n] = s;
    }
    __syncthreads();

    // ---------------- Phase 2: out = target + Y @ W^T ----------------
    // A = Y (16x16, K split 4x4) is invariant across d-blocks: hoist the loads.
    v2f a0, a1, a2, a3;
    {
        const float* yrow = yS + l16 * YS;
        a0.x = yrow[klo];      a0.y = yrow[klo + 1];
        a1.x = yrow[4 + klo];  a1.y = yrow[5 + klo];
        a2.x = yrow[8 + klo];  a2.y = yrow[9 + klo];
        a3.x = yrow[12 + klo]; a3.y = yrow[13 + klo];
    }

    for (int d0 = (wave << 4); d0 < D_DIM; d0 += (NWAVES << 4)) {
        // C = cached target block from LDS (disjoint 16-bank halves -> conflict-free)
        v8f c;
        #pragma unroll
        for (int i = 0; i < 8; ++i)
            c[i] = tgtS[(i + (half << 3)) * TS + d0 + l16];

        // B = (W^T) block: B[K][N] = W[d0+N][K]; aligned float2 from one W row per lane
        const float* wrow = wgt + (long)(d0 + l16) * 16;
        const v2f b0 = *(const v2f*)(wrow + 0  + klo);
        const v2f b1 = *(const v2f*)(wrow + 4  + klo);
        const v2f b2 = *(const v2f*)(wrow + 8  + klo);
        const v2f b3 = *(const v2f*)(wrow + 12 + klo);

        c = __builtin_amdgcn_wmma_f32_16x16x4_f32(false, a0, false, b0, (short)0, c, false, false);
        c = __builtin_amdgcn_wmma_f32_16x16x4_f32(false, a1, false, b1, (short)0, c, false, false);
        c = __builtin_amdgcn_wmma_f32_16x16x4_f32(false, a2, false, b2, (short)0, c, false, false);
        c = __builtin_amdgcn_wmma_f32_16x16x4_f32(false, a3, false, b3, (short)0, c, false, false);

        // Coalesced 64B stores per 16-lane half.
        #pragma unroll
        for (int i = 0; i < 8; ++i)
            outT[(long)(i + (half << 3)) * D_DIM + d0 + l16] = c[i];
    }
}

extern "C" void kernel_launch(void* const* d_in, const int* in_sizes, int n_in,
                              void* d_out, int out_size, void* d_ws, size_t ws_size,
                              hipStream_t stream) {
    (void)n_in; (void)out_size; (void)d_ws; (void)ws_size;
    const float* src = (const float*)d_in[0];
    const float* tgt = (const float*)d_in[1];
    const float* wgt = (const float*)d_in[2];
    float* out = (float*)d_out;

    const int rows = in_sizes[0] / D_DIM;     // B*S = 16384
    const int grid = rows / TILE_M;           // 1024 workgroups
    const size_t smem_bytes =
        (size_t)(TILE_M * TS + 2 * TILE_M * SSY + TILE_M * YS + NWAVES * 256) * sizeof(float);

    // CDNA5 allows up to 320KB LDS per workgroup; raise the dynamic-LDS cap.
    hipFuncSetAttribute(reinterpret_cast<const void*>(lrproj_fused),
                        hipFuncAttributeMaxDynamicSharedMemorySize,
                        (int)smem_bytes);

    lrproj_fused<<<grid, 128, smem_bytes, stream>>>(src, tgt, wgt, out);
}